// DFIREScoreModule_61383672594451
// MI455X (gfx1250) — compile-verified
//
#include <hip/hip_runtime.h>

typedef float v2f __attribute__((ext_vector_type(2)));
typedef float v8f __attribute__((ext_vector_type(8)));

#define TILE  16
#define NTYPE 85
#define NBIN  28

// Wave handles a contiguous chunk of upper-tri 16x16 tile pairs (row-major:
// ti fixed, tj ascending), so i-tile state is reused across inner iterations.
// All tile-walk state is wave-uniform (SGPRs) via readfirstlane.
//
// dist^2(m,n) = |xi_m|^2 + |xj_n|^2 - 2 xi_m . xj_n
// WMMA A[m] = (-2x_i, -2y_i, -2z_i, |xi|^2), B[:,n] = (x_j, y_j, z_j, 1)
//   => C[m][n] = |xi_m|^2 - 2 xi_m.xj_n ; then add per-lane |xj_n|^2.
//
// i<j as one integer compare:  m < (tj-ti)*16 + l16  (thr per tile, 1 v_cmp/elem)
// LUT lerp: single b64 gather of the adjacent bin pair {pot[k], pot[k+1]},
// k = min(d0,26), alpha clamped to 1 reproduces the d0==27 edge exactly.
__global__ __launch_bounds__(256) void dfire_pair_kernel(
    const float* __restrict__ coords,
    const float* __restrict__ pot,
    const int* __restrict__ res_ids,
    const int* __restrict__ type_idx,
    int nAtoms, int M, int T, int chunk,
    float* __restrict__ blockPartials)
{
  const int lane = threadIdx.x & 31;
  const int half = lane >> 4;       // which 16-lane half of the wave
  const int l16  = lane & 15;
  const int wave  = __builtin_amdgcn_readfirstlane(threadIdx.x >> 5);
  const int gwave = blockIdx.x * 8 + wave;

  const float CUT2 = 19.6f * 19.6f;               // compare on d^2, sqrt only if taken

  // per-lane C/D row index for each accumulator register: m = rr + 8*half
  int mRow[8];
  #pragma unroll
  for (int rr = 0; rr < 8; ++rr) mRow[rr] = rr + (half << 3);

  float sum = 0.0f;

  int p0 = gwave * chunk;
  int p1 = p0 + chunk; if (p1 > T) p1 = T;

  if (p0 < p1) {
    // ---- decode p0 -> (ti, tj); row-major over the upper triangle (all SALU)
    int q = T - 1 - p0;
    int r = (int)((sqrtf((float)(8 * q + 1)) - 1.0f) * 0.5f);
    while ((r + 1) * (r + 2) / 2 <= q) ++r;  // integer fixups for fp rounding
    while (r * (r + 1) / 2 > q) --r;
    int ti = M - 1 - r;
    int tj = M - 1 - (q - r * (r + 1) / 2);

    // ---- per-i-tile state (reloaded only when ti changes)
    v2f a;
    unsigned rowBase[8];   // t1 * NTYPE * NBIN per C/D row of this half
    int      resRow[8];    // res_id per C/D row

    auto loadITile = [&]() {
      int iaS = ti * TILE + l16;
      unsigned ia = (unsigned)((iaS < nAtoms) ? iaS : (nAtoms - 1));
      float xi = coords[ia * 3u + 0u];
      float yi = coords[ia * 3u + 1u];
      float zi = coords[ia * 3u + 2u];
      int pk_i = (res_ids[ia] << 8) | (type_idx[ia] & 0xFF);
      float ni2 = xi * xi + yi * yi + zi * zi;
      // A-matrix 16x4 f32 layout: lanes 0-15 hold K=0,1 ; lanes 16-31 hold K=2,3
      a[0] = half ? (-2.0f * zi) : (-2.0f * xi);
      a[1] = half ? ni2          : (-2.0f * yi);
      #pragma unroll
      for (int rr = 0; rr < 8; ++rr) {
        int pk = __shfl(pk_i, mRow[rr], 32); // row broadcast, hoisted out of tj loop
        rowBase[rr] = (unsigned)(pk & 0xFF) * (NTYPE * NBIN);
        resRow[rr]  = pk >> 8;
      }
    };
    loadITile();

    // ---- prime j-tile pipeline (current tile's raw data)
    int ja0S = tj * TILE + l16;
    unsigned ja0 = (unsigned)((ja0S < nAtoms) ? ja0S : (nAtoms - 1));
    float xj = coords[ja0 * 3u + 0u];
    float yj = coords[ja0 * 3u + 1u];
    float zj = coords[ja0 * 3u + 2u];
    int res_j = res_ids[ja0];
    int t2    = type_idx[ja0];

    for (int p = p0; p < p1; ++p) {
      int jglob = tj * TILE + l16;
      bool jvalid = (jglob < nAtoms) & (t2 >= 0);
      int  thr = (tj - ti) * TILE + l16;              // i<j threshold for this tile
      float nj2 = xj * xj + yj * yj + zj * zj;
      unsigned colBase = (unsigned)t2 * NBIN;

      v2f b;
      b[0] = half ? zj   : xj;
      b[1] = half ? 1.0f : yj;

      v8f c = {};
      c = __builtin_amdgcn_wmma_f32_16x16x4_f32(false, a, false, b,
                                                (short)0, c, false, false);

      // ---- software-pipeline: issue next j-tile loads behind the epilogue
      int tjn = (tj + 1 < M) ? (tj + 1) : (ti + 1);   // scalar select
      if (tjn > M - 1) tjn = M - 1;                   // harmless in-bounds prefetch
      int janS = tjn * TILE + l16;
      unsigned jan = (unsigned)((janS < nAtoms) ? janS : (nAtoms - 1));
      float xjn = coords[jan * 3u + 0u];
      float yjn = coords[jan * 3u + 1u];
      float zjn = coords[jan * 3u + 2u];
      int res_jn = res_ids[jan];
      int t2n    = type_idx[jan];

      // ---- single epilogue: every predicate is a native compare -> lane mask
      #pragma unroll
      for (int rr = 0; rr < 8; ++rr) {
        int sep = resRow[rr] - res_j; sep = (sep < 0) ? -sep : sep;
        float d2 = c[rr] + nj2;
        bool ok = (sep > 2) & (d2 < CUT2) & (mRow[rr] < thr) & jvalid;
        if (ok) {                                     // EXEC-masked region
          // |d2| clamp folds into a v_sqrt source modifier
          float d  = __builtin_amdgcn_sqrtf(fabsf(d2));
          float ds = fmaf(d, 1.0f / 0.7f, 1e-8f / 0.7f);
          // in-branch d < 19.6 => ds < 28, so no 27-clamp needed before floor
          float d0f = floorf(ds);
          float kf  = fminf(d0f, 26.0f);
          float alpha = fminf(ds - kf, 1.0f);         // ==1 at the d0==27 edge
          unsigned k = (unsigned)(int)kf;
          unsigned base = rowBase[rr] + colBase + k;  // 32-bit unsigned addressing
          // one b64 gather of adjacent bins (4B-aligned; DWORD mode handles it)
          float2 ee = *reinterpret_cast<const float2*>(pot + base);
          sum += ee.x + alpha * (ee.y - ee.x);
        }
      }

      // ---- rotate pipeline and advance row-major position (scalar)
      xj = xjn; yj = yjn; zj = zjn; res_j = res_jn; t2 = t2n;
      if (++tj == M) {
        ++ti;
        tj = ti;
        if (p + 1 < p1) loadITile();
      }
    }
  }

  // ---- deterministic reduction: wave shuffle -> block LDS -> d_ws partial
  for (int off = 16; off > 0; off >>= 1) sum += __shfl_xor(sum, off, 32);
  __shared__ float lds[8];
  if (lane == 0) lds[wave] = sum;
  __syncthreads();
  if (threadIdx.x == 0) {
    float t = 0.0f;
    for (int w = 0; w < 8; ++w) t += lds[w];
    blockPartials[blockIdx.x] = t;
  }
}

__global__ __launch_bounds__(256) void dfire_reduce_kernel(
    const float* __restrict__ part, int n, float* __restrict__ out)
{
  float s = 0.0f;
  for (int i = threadIdx.x; i < n; i += blockDim.x) s += part[i];
  for (int off = 16; off > 0; off >>= 1) s += __shfl_xor(s, off, 32);
  __shared__ float lds[8];
  int lane = threadIdx.x & 31, wave = threadIdx.x >> 5;
  if (lane == 0) lds[wave] = s;
  __syncthreads();
  if (threadIdx.x == 0) {
    float t = 0.0f;
    for (int w = 0; w < 8; ++w) t += lds[w];
    out[0] = t;
  }
}

extern "C" void kernel_launch(void* const* d_in, const int* in_sizes, int n_in,
                              void* d_out, int out_size, void* d_ws, size_t ws_size,
                              hipStream_t stream) {
  const float* coords = (const float*)d_in[0];   // [N,3] f32
  const float* pot    = (const float*)d_in[1];   // [85,85,28] f32
  const int*   res    = (const int*)d_in[2];     // [N] i32
  const int*   typ    = (const int*)d_in[3];     // [N] i32
  // d_in[4]/d_in[5] (i_idx/j_idx) intentionally unused: pairs are regenerated
  // as 16x16 tiles, avoiding 268 MB of index streaming.
  (void)n_in; (void)out_size;

  int nAtoms = in_sizes[0] / 3;
  int M = (nAtoms + TILE - 1) / TILE;
  int T = M * (M + 1) / 2;

  int blocks = 2048;                              // 16384 waves
  int maxBlocks = (int)(ws_size / sizeof(float));
  if (blocks > maxBlocks) blocks = maxBlocks;
  if (blocks > T) blocks = (T > 0) ? T : 1;
  if (blocks < 1) blocks = 1;
  int totalWaves = blocks * 8;
  int chunk = (T + totalWaves - 1) / totalWaves;  // contiguous tile-pair chunk/wave

  dfire_pair_kernel<<<blocks, 256, 0, stream>>>(coords, pot, res, typ,
                                                nAtoms, M, T, chunk,
                                                (float*)d_ws);
  dfire_reduce_kernel<<<1, 256, 0, stream>>>((const float*)d_ws, blocks,
                                             (float*)d_out);
}